// TanhAttention_12360915878884
// MI455X (gfx1250) — compile-verified
//
#include <hip/hip_runtime.h>
#include <math.h>

#define D      256
#define TX     512
#define TM     512
#define BATCH  4

typedef __attribute__((ext_vector_type(2))) float v2f;
typedef __attribute__((ext_vector_type(8))) float v8f;

__device__ __forceinline__ float fast_tanh(float x) {
#if __has_builtin(__builtin_amdgcn_tanhf)
  return __builtin_amdgcn_tanhf(x);
#else
  float r;
  asm("v_tanh_f32 %0, %1" : "=v"(r) : "v"(x));
  return r;
#endif
}

__device__ __forceinline__ float fast_exp(float x) {
#if __has_builtin(__builtin_amdgcn_exp2f)
  return __builtin_amdgcn_exp2f(x * 1.4426950408889634f);
#else
  return __expf(x);
#endif
}

// Out[r][e] = sum_d In[r][d] * W[e][d] (+ bias[e])
// A = In (M x K), B[k][n] = W[n][k]  -> both fragments are contiguous v2f loads.
// Each wave computes two adjacent 16x16 N-tiles (shares A fragment, breaks the
// WMMA accumulator dependency chain).
__global__ __launch_bounds__(128) void proj_gemm_kernel(
    const float* __restrict__ In, const float* __restrict__ W,
    const float* __restrict__ bias, float* __restrict__ Out, int Mrows) {
  const int warp = threadIdx.x >> 5;
  const int lane = threadIdx.x & 31;
  const int half = lane >> 4;       // 0: K pair {0,1}, 1: K pair {2,3}
  const int l    = lane & 15;       // M index (A) / N index (B)

  const int pairsPerRow = D / 32;   // 8 dual-tile columns
  int tile = blockIdx.x * (blockDim.x >> 5) + warp;
  int mt = tile / pairsPerRow;
  int np = tile % pairsPerRow;
  if (mt * 16 >= Mrows) return;
  const int row0 = mt * 16;
  const int col0 = np * 32;

  v8f acc0 = {}; v8f acc1 = {};
  const float* arow  = In + (size_t)(row0 + l) * D;
  const float* b0row = W  + (size_t)(col0 + l) * D;
  const float* b1row = W  + (size_t)(col0 + 16 + l) * D;
#pragma unroll 8
  for (int k = 0; k < D; k += 4) {
    v2f a  = *(const v2f*)(arow  + k + 2 * half);
    v2f b0 = *(const v2f*)(b0row + k + 2 * half);
    v2f b1 = *(const v2f*)(b1row + k + 2 * half);
    acc0 = __builtin_amdgcn_wmma_f32_16x16x4_f32(false, a, false, b0, (short)0, acc0, false, false);
    acc1 = __builtin_amdgcn_wmma_f32_16x16x4_f32(false, a, false, b1, (short)0, acc1, false, false);
  }
  float bv0 = 0.0f, bv1 = 0.0f;
  if (bias) { bv0 = bias[col0 + l]; bv1 = bias[col0 + 16 + l]; }
#pragma unroll
  for (int j = 0; j < 8; ++j) {
    int r = row0 + j + 8 * half;            // C/D layout: VGPR j -> M=j (lanes 0-15), M=8+j (16-31)
    Out[(size_t)r * D + col0 + l]      = acc0[j] + bv0;
    Out[(size_t)r * D + col0 + 16 + l] = acc1[j] + bv1;
  }
}

// One block per (b,x) row: scores + mask + softmax, S written to output.
__global__ __launch_bounds__(256) void scores_softmax_kernel(
    const float* __restrict__ item1, const float* __restrict__ item2,
    const float* __restrict__ wst, const int* __restrict__ mask,
    float* __restrict__ Sout) {
  __shared__ float sc[TM];
  __shared__ float red[256];

  const int tid  = threadIdx.x;
  const int b    = blockIdx.x / TX;
  const int x    = blockIdx.x % TX;
  const int warp = tid >> 5;
  const int lane = tid & 31;

  // Preload this lane's 8 item1-row values and 8 weights into registers.
  const size_t rb = ((size_t)b * TX + x) * D;
  const float4 f10 = *(const float4*)(item1 + rb + lane * 8);
  const float4 f11 = *(const float4*)(item1 + rb + lane * 8 + 4);
  const float4 w0  = *(const float4*)(wst + lane * 8);
  const float4 w1  = *(const float4*)(wst + lane * 8 + 4);

  const float* i2b = item2 + (size_t)b * TM * D;
  for (int m = warp; m < TM; m += 8) {
    const float* row = i2b + (size_t)m * D + lane * 8;
    if (m + 8 < TM) __builtin_prefetch(row + 8 * D, 0, 3);
    float4 r0 = *(const float4*)(row);
    float4 r1 = *(const float4*)(row + 4);
    float p;
    p  = fast_tanh(f10.x + r0.x) * w0.x;
    p += fast_tanh(f10.y + r0.y) * w0.y;
    p += fast_tanh(f10.z + r0.z) * w0.z;
    p += fast_tanh(f10.w + r0.w) * w0.w;
    p += fast_tanh(f11.x + r1.x) * w1.x;
    p += fast_tanh(f11.y + r1.y) * w1.y;
    p += fast_tanh(f11.z + r1.z) * w1.z;
    p += fast_tanh(f11.w + r1.w) * w1.w;
    // wave32 reduction
    p += __shfl_xor(p, 16, 32);
    p += __shfl_xor(p, 8, 32);
    p += __shfl_xor(p, 4, 32);
    p += __shfl_xor(p, 2, 32);
    p += __shfl_xor(p, 1, 32);
    if (lane == 0) sc[m] = p;
  }
  __syncthreads();

  // mask + softmax over TM (each thread owns m = tid and m = tid+256)
  float s0 = sc[tid];
  float s1 = sc[tid + 256];
  if (mask[b * TM + tid] == 0)       s0 = -INFINITY;
  if (mask[b * TM + tid + 256] == 0) s1 = -INFINITY;

  red[tid] = fmaxf(s0, s1);
  __syncthreads();
  for (int s = 128; s > 0; s >>= 1) {
    if (tid < s) red[tid] = fmaxf(red[tid], red[tid + s]);
    __syncthreads();
  }
  const float rowmax = red[0];
  __syncthreads();

  const float e0 = fast_exp(s0 - rowmax);
  const float e1 = fast_exp(s1 - rowmax);
  red[tid] = e0 + e1;
  __syncthreads();
  for (int s = 128; s > 0; s >>= 1) {
    if (tid < s) red[tid] += red[tid + s];
    __syncthreads();
  }
  const float inv = 1.0f / red[0];

  const size_t base = ((size_t)b * TX + x) * TM;
  Sout[base + tid]       = e0 * inv;
  Sout[base + 256 + tid] = e1 * inv;
}

// out[b] = S[b] (TX x TM) @ memory[b] (TM x D), f32 WMMA, dual N-tiles per wave.
__global__ __launch_bounds__(128) void attn_out_kernel(
    const float* __restrict__ S, const float* __restrict__ Mem,
    float* __restrict__ Out) {
  const int warp = threadIdx.x >> 5;
  const int lane = threadIdx.x & 31;
  const int half = lane >> 4;
  const int l    = lane & 15;

  const int pairsPerRow   = D / 32;                 // 8
  const int tilesPerBatch = (TX / 16) * pairsPerRow; // 256
  int tile  = blockIdx.x * (blockDim.x >> 5) + warp;
  int batch = tile / tilesPerBatch;
  int t     = tile % tilesPerBatch;
  int mt = t / pairsPerRow;
  int np = t % pairsPerRow;
  const int row0 = mt * 16;
  const int col0 = np * 32;

  const float* Sb = S   + (size_t)batch * TX * TM;
  const float* Mb = Mem + (size_t)batch * TM * D;
  const float* arow = Sb + (size_t)(row0 + l) * TM;

  v8f acc0 = {}; v8f acc1 = {};
#pragma unroll 4
  for (int k = 0; k < TM; k += 4) {
    const int kr = k + 2 * half;
    v2f a = *(const v2f*)(arow + kr);
    v2f b0, b1;
    b0.x = Mb[(size_t)kr * D + col0 + l];
    b0.y = Mb[(size_t)(kr + 1) * D + col0 + l];
    b1.x = Mb[(size_t)kr * D + col0 + 16 + l];
    b1.y = Mb[(size_t)(kr + 1) * D + col0 + 16 + l];
    acc0 = __builtin_amdgcn_wmma_f32_16x16x4_f32(false, a, false, b0, (short)0, acc0, false, false);
    acc1 = __builtin_amdgcn_wmma_f32_16x16x4_f32(false, a, false, b1, (short)0, acc1, false, false);
  }
#pragma unroll
  for (int j = 0; j < 8; ++j) {
    int r = row0 + j + 8 * half;
    Out[((size_t)batch * TX + r) * D + col0 + l]      = acc0[j];
    Out[((size_t)batch * TX + r) * D + col0 + 16 + l] = acc1[j];
  }
}

extern "C" void kernel_launch(void* const* d_in, const int* in_sizes, int n_in,
                              void* d_out, int out_size, void* d_ws, size_t ws_size,
                              hipStream_t stream) {
  const float* x    = (const float*)d_in[0];
  const float* mem  = (const float*)d_in[1];
  const int*   mask = (const int*)d_in[2];
  const float* w1   = (const float*)d_in[3];
  const float* b1   = (const float*)d_in[4];
  const float* w2   = (const float*)d_in[5];
  const float* wst  = (const float*)d_in[6];

  float* item1 = (float*)d_ws;                          // 2048*256 f32
  float* item2 = item1 + (size_t)BATCH * TX * D;        // 2048*256 f32
  float* out   = (float*)d_out;                         // [B,Tx,D]
  float* Sout  = out + (size_t)BATCH * TX * D;          // [B,Tx,Tm]

  const int rowsX = BATCH * TX;   // 2048
  const int rowsM = BATCH * TM;   // 2048
  dim3 blk(128);

  int tilesProj = (rowsX / 16) * (D / 32);              // 1024 dual-tiles
  proj_gemm_kernel<<<tilesProj / 4, blk, 0, stream>>>(x,   w1, b1,      item1, rowsX);
  proj_gemm_kernel<<<tilesProj / 4, blk, 0, stream>>>(mem, w2, nullptr, item2, rowsM);

  scores_softmax_kernel<<<BATCH * TX, 256, 0, stream>>>(item1, item2, wst, mask, Sout);

  int tilesOut = BATCH * (TX / 16) * (D / 32);          // 1024 dual-tiles
  attn_out_kernel<<<tilesOut / 4, blk, 0, stream>>>(Sout, mem, out);
}